// VGGLikeSNN_26104811225511
// MI455X (gfx1250) — compile-verified
//
#include <hip/hip_runtime.h>

// ---------------------------------------------------------------------------
// CDNA5 (gfx1250) spiking-VGG forward pass.
//  - Matrix ops: V_WMMA_F32_16X16X4_F32 (native f32 WMMA -> exact precision
//    parity with the f32 reference). Wave32; layouts per ISA 7.12.2.
//  - Input staging: GLOBAL_LOAD_ASYNC_TO_LDS (ASYNCcnt) via the gfx1250
//    builtins (parameter types per hipcc diagnostics: AS1 int*/int4* source,
//    AS3 destination), else plain load+ds_store fallback.
//  - Conv kernels templated on (CIN, F) so all im2col addressing is
//    compile-time affine (no runtime div/mod in the hot loop).
// ---------------------------------------------------------------------------

typedef float v2f __attribute__((ext_vector_type(2)));
typedef float v4f __attribute__((ext_vector_type(4)));
typedef float v8f __attribute__((ext_vector_type(8)));

__device__ __forceinline__ v8f wmma_f32_16x16x4(v2f a, v2f b, v8f c) {
  // 8-arg form: (neg_a, A, neg_b, B, c_mod, C, reuse_a, reuse_b)
  return __builtin_amdgcn_wmma_f32_16x16x4_f32(false, a, false, b,
                                               (short)0, c, false, false);
}

#if __has_builtin(__builtin_amdgcn_global_load_async_to_lds_b128) && \
    __has_builtin(__builtin_amdgcn_global_load_async_to_lds_b32)  && \
    __has_builtin(__builtin_amdgcn_s_wait_asynccnt)
#define USE_ASYNC_LDS 1
typedef int vi4 __attribute__((vector_size(16)));           // GCC-style int4
typedef __attribute__((address_space(1))) int gi32;         // global int
typedef __attribute__((address_space(3))) int li32;         // LDS int
typedef __attribute__((address_space(1))) vi4 gvi4;         // global int4
typedef __attribute__((address_space(3))) vi4 lvi4;         // LDS int4
#else
#define USE_ASYNC_LDS 0
#endif

// ---------------------------------------------------------------------------
// Spiking 3x3 conv: out = min((p - t_min) @ W + (t_max - D[region]), t_max)
// Block: 256 threads = 8 waves, 2 (pixel-tile) x 4 (channel-tile).
// Covers 32 pixels along W and 64 output channels at one (b, h).
// LDS holds the raw [3][34][CIN] input window (pad cells = t_min, so the
// "- t_min" at A-fetch time yields 0 for padding, matching the reference).
// ---------------------------------------------------------------------------
template <int CIN, int F>
__global__ __launch_bounds__(256)
void snn_conv3x3(const float* __restrict__ in,   // [B,H,W,CIN] spike times
                 const float* __restrict__ Wk,   // [9*CIN, F]
                 const float* __restrict__ D,    // [9, F] region thresholds
                 float* __restrict__ out,        // [B,H,W,F]
                 int H, int Wd, float t_min, float t_max)
{
  __shared__ float lds[3 * 34 * CIN];

  const int wtiles = Wd >> 5;
  const int wt = blockIdx.x % wtiles;
  const int ft = blockIdx.x / wtiles;
  const int h  = blockIdx.y;
  const int b  = blockIdx.z;
  const int w0 = wt * 32;
  const int f0 = ft * 64;

  // ---- stage the (h-1..h+1) x (w0-1..w0+32) x CIN window into LDS ----
  if constexpr (CIN == 1) {
    for (int i = threadIdx.x; i < 3 * 34; i += 256) {
      const int col = i % 34, r = i / 34;
      const int hr = h - 1 + r;
      const int wc = w0 - 1 + col;
      const bool ib = (hr >= 0 && hr < H && wc >= 0 && wc < Wd);
      const float* gp = in + ((size_t)b * H + hr) * Wd + wc;
#if USE_ASYNC_LDS
      if (ib)
        __builtin_amdgcn_global_load_async_to_lds_b32((gi32*)gp,
                                                      (li32*)&lds[i], 0, 0);
      else
        lds[i] = t_min;
#else
      lds[i] = ib ? *gp : t_min;
#endif
    }
  } else {
    constexpr int CQ = CIN / 4;          // quads per window cell
    for (int i = threadIdx.x; i < 3 * 34 * CQ; i += 256) {
      const int c4  = i % CQ;
      const int t   = i / CQ;            // r*34 + col
      const int col = t % 34, r = t / 34;
      const int hr  = h - 1 + r;
      const int wc  = w0 - 1 + col;
      const bool ib = (hr >= 0 && hr < H && wc >= 0 && wc < Wd);
      float* lp = &lds[t * CIN + c4 * 4];
      const float* gp = in + (((size_t)b * H + hr) * Wd + wc) * CIN + c4 * 4;
#if USE_ASYNC_LDS
      if (ib) {
        __builtin_amdgcn_global_load_async_to_lds_b128((gvi4*)gp,
                                                       (lvi4*)lp, 0, 0);
      } else {
        v4f pad = {t_min, t_min, t_min, t_min};
        *(v4f*)lp = pad;
      }
#else
      v4f val;
      if (ib) val = *(const v4f*)gp;
      else    val = (v4f){t_min, t_min, t_min, t_min};
      *(v4f*)lp = val;
#endif
    }
  }
#if USE_ASYNC_LDS
  __builtin_amdgcn_s_wait_asynccnt(0);
#endif
  __syncthreads();

  // ---- wave / lane decomposition (A 16x4 f32 layout, ISA 7.12.2) ----
  const int tid   = threadIdx.x;
  const int lane  = tid & 31;
  const int wave  = tid >> 5;
  const int waveM = wave & 1;          // pixel sub-tile (16 px)
  const int waveN = wave >> 1;         // channel sub-tile (16 ch)
  const int l15   = lane & 15;
  const int hi    = lane >> 4;         // lanes 16..31 hold K+2 / K+3
  const int koff  = hi * 2;

  const int mBase = waveM * 16;        // pixel base within 32-px window
  const int n     = f0 + waveN * 16 + l15;
  const int mcol  = mBase + l15;       // A-row's window column base

  v8f acc = {0.f, 0.f, 0.f, 0.f, 0.f, 0.f, 0.f, 0.f};

  if constexpr (CIN == 1) {
    // K = 9, padded to 12. Steps k0 = 0, 4 are fully in-range.
#pragma unroll
    for (int k0 = 0; k0 < 8; k0 += 4) {
      const int ka = k0 + koff;
      const int kb = ka + 1;
      v2f a  = { lds[(ka / 3) * 34 + mcol + (ka % 3)] - t_min,
                 lds[(kb / 3) * 34 + mcol + (kb % 3)] - t_min };
      v2f bb = { Wk[(size_t)ka * F + n], Wk[(size_t)kb * F + n] };
      acc = wmma_f32_16x16x4(a, bb, acc);
    }
    // Tail step k0 = 8: only lanes 0..15, first element (k=8) are valid.
    {
      const float a0 = (koff == 0) ? (lds[2 * 34 + mcol + 2] - t_min) : 0.f;
      const float b0 = (koff == 0) ? Wk[(size_t)8 * F + n] : 0.f;
      v2f a  = { a0, 0.f };
      v2f bb = { b0, 0.f };
      acc = wmma_f32_16x16x4(a, bb, acc);
    }
  } else {
    // K = 9*CIN; steps of 4 never cross a (dy,dx) patch boundary.
#pragma unroll
    for (int p = 0; p < 9; ++p) {
      constexpr int C = CIN;
      const int dy = p / 3, dx = p % 3;
      const float* arow = &lds[(dy * 34 + mcol + dx) * C + koff];
      const float* brow = Wk + ((size_t)p * C + koff) * F + n;
#pragma unroll 4
      for (int c0 = 0; c0 < C; c0 += 4) {
        v2f a  = { arow[c0] - t_min, arow[c0 + 1] - t_min };
        v2f bb = { brow[(size_t)c0 * F], brow[(size_t)(c0 + 1) * F] };
        acc = wmma_f32_16x16x4(a, bb, acc);
      }
    }
  }

  // ---- epilogue: border-aware threshold + clamp ----
  // C/D layout: VGPR v holds row M = v + 8*hi, column N = l15.
#pragma unroll
  for (int v = 0; v < 8; ++v) {
    const int mLocal = v + hi * 8;
    const int wpix   = w0 + mBase + mLocal;
    const bool top = (h == 0), bot = (h == H - 1);
    const bool lef = (wpix == 0), rig = (wpix == Wd - 1);
    int reg;
    if (top)      reg = lef ? 1 : (rig ? 3 : 2);
    else if (bot) reg = lef ? 7 : (rig ? 5 : 6);
    else if (lef) reg = 8;
    else if (rig) reg = 4;
    else          reg = 0;
    const float Dv  = D[reg * F + n];
    const float val = fminf(acc[v] + (t_max - Dv), t_max);
    out[(((size_t)b * H + h) * Wd + wpix) * F + n] = val;
  }
}

// ---------------------------------------------------------------------------
// 2x2 max-pool, NHWC.
// ---------------------------------------------------------------------------
__global__ void snn_pool2(const float* __restrict__ in,
                          float* __restrict__ out,
                          int H, int Wd, int C, int n)
{
  int i = blockIdx.x * blockDim.x + threadIdx.x;
  if (i >= n) return;
  int c = i % C;
  int t = i / C;
  const int Wo = Wd >> 1, Ho = H >> 1;
  int w = t % Wo; t /= Wo;
  int h = t % Ho;
  int b = t / Ho;
  size_t base = (((size_t)b * H + 2 * h) * Wd + 2 * w) * C + c;
  size_t rs   = (size_t)Wd * C;
  float m0 = fmaxf(in[base], in[base + C]);
  float m1 = fmaxf(in[base + rs], in[base + rs + C]);
  out[i] = fmaxf(m0, m1);
}

// ---------------------------------------------------------------------------
// Dense layer GEMM, split-K: partial (x - t_min) @ W accumulated via
// global f32 atomics. One wave per block; M=32 (batch) x N=256, K=32768.
// ---------------------------------------------------------------------------
__global__ __launch_bounds__(32)
void snn_dense_splitk(const float* __restrict__ x,   // [32, Kdim]
                      const float* __restrict__ Wd,  // [Kdim, N]
                      float* __restrict__ accbuf,    // [32, N] (pre-zeroed)
                      int Kdim, int N, float t_min)
{
  const int mt   = blockIdx.x & 1;
  const int nt   = blockIdx.x >> 1;
  const int lane = threadIdx.x & 31;
  const int l15  = lane & 15;
  const int hi   = lane >> 4;
  const int koff = hi * 2;

  const int kchunk = Kdim / gridDim.y;
  const int kb     = blockIdx.y * kchunk;

  const int m = mt * 16 + l15;
  const int n = nt * 16 + l15;
  const float* xr = x  + (size_t)m * Kdim + koff;
  const float* wr = Wd + (size_t)koff * N + n;

  v8f acc = {0.f, 0.f, 0.f, 0.f, 0.f, 0.f, 0.f, 0.f};
  for (int k0 = kb; k0 < kb + kchunk; k0 += 4) {
    v2f a  = { xr[k0] - t_min, xr[k0 + 1] - t_min };
    v2f bb = { wr[(size_t)k0 * N], wr[(size_t)(k0 + 1) * N] };
    acc = wmma_f32_16x16x4(a, bb, acc);
  }
#pragma unroll
  for (int v = 0; v < 8; ++v) {
    const int row = mt * 16 + v + hi * 8;
    atomicAdd(&accbuf[(size_t)row * N + n], acc[v]);
  }
}

__global__ void snn_zero(float* __restrict__ p, int n)
{
  int i = blockIdx.x * blockDim.x + threadIdx.x;
  if (i < n) p[i] = 0.0f;
}

__global__ void snn_dense_finalize(const float* __restrict__ acc,
                                   const float* __restrict__ Dd1,
                                   float* __restrict__ outd,
                                   float t_max, int n)
{
  int i = blockIdx.x * blockDim.x + threadIdx.x;
  if (i < n) {
    int c = i & 255;
    outd[i] = fminf(acc[i] + (t_max - Dd1[c]), t_max);
  }
}

// out[b,j] = Dout[j] + sum_k (7500 - dense[b,k]) * Wout[k,j]
__global__ void snn_out_layer(const float* __restrict__ dense,
                              const float* __restrict__ Wout,
                              const float* __restrict__ Dout,
                              float* __restrict__ out)
{
  int t = threadIdx.x;          // 320 threads = 32 batch x 10 classes
  int b = t / 10, j = t % 10;
  float s = Dout[j];
  for (int k = 0; k < 256; ++k)
    s += (7500.0f - dense[b * 256 + k]) * Wout[k * 10 + j];
  out[b * 10 + j] = s;
}

// ---------------------------------------------------------------------------
// Host-side launch sequence (graph-capture safe: kernels only, one stream).
// Workspace plan (two 32 MiB ping-pong regions, 64 MiB total):
//   bufA = ws        bufB = ws + 32MiB
//   conv1: x    -> bufA [32,64,64,64]
//   conv2: bufA -> bufB
//   pool1: bufB -> bufA [32,32,32,64]
//   conv3: bufA -> bufB [32,32,32,128]
//   conv4: bufB -> bufA
//   pool2: bufA -> bufB [32,16,16,128] = [32,32768]
//   dense acc -> bufA[0:8192], dense out -> bufA + 16384 floats
// ---------------------------------------------------------------------------
extern "C" void kernel_launch(void* const* d_in, const int* in_sizes, int n_in,
                              void* d_out, int out_size, void* d_ws, size_t ws_size,
                              hipStream_t stream)
{
  (void)in_sizes; (void)n_in; (void)out_size; (void)ws_size;

  const float* x    = (const float*)d_in[0];
  const float* Wc1  = (const float*)d_in[1];
  const float* Dc1  = (const float*)d_in[2];
  const float* Wc2  = (const float*)d_in[3];
  const float* Dc2  = (const float*)d_in[4];
  const float* Wc3  = (const float*)d_in[5];
  const float* Dc3  = (const float*)d_in[6];
  const float* Wc4  = (const float*)d_in[7];
  const float* Dc4  = (const float*)d_in[8];
  const float* Wd1  = (const float*)d_in[9];
  const float* Dd1  = (const float*)d_in[10];
  const float* Wout = (const float*)d_in[11];
  const float* Dout = (const float*)d_in[12];

  char*  ws   = (char*)d_ws;
  float* bufA = (float*)ws;
  float* bufB = (float*)(ws + (size_t)33554432);   // 32 MiB

  // conv1: [32,64,64,1] -> [32,64,64,64]
  snn_conv3x3<1, 64><<<dim3(2, 64, 32), 256, 0, stream>>>(
      x, Wc1, Dc1, bufA, 64, 64, 0.0f, 1500.0f);
  // conv2: 64 -> 64
  snn_conv3x3<64, 64><<<dim3(2, 64, 32), 256, 0, stream>>>(
      bufA, Wc2, Dc2, bufB, 64, 64, 1500.0f, 3000.0f);
  // pool1: 64x64 -> 32x32
  {
    int n = 32 * 32 * 32 * 64;
    snn_pool2<<<(n + 255) / 256, 256, 0, stream>>>(bufB, bufA, 64, 64, 64, n);
  }
  // conv3: 64 -> 128
  snn_conv3x3<64, 128><<<dim3(2, 32, 32), 256, 0, stream>>>(
      bufA, Wc3, Dc3, bufB, 32, 32, 3000.0f, 4500.0f);
  // conv4: 128 -> 128
  snn_conv3x3<128, 128><<<dim3(2, 32, 32), 256, 0, stream>>>(
      bufB, Wc4, Dc4, bufA, 32, 32, 4500.0f, 6000.0f);
  // pool2: 32x32 -> 16x16, flatten to [32, 32768]
  {
    int n = 32 * 16 * 16 * 128;
    snn_pool2<<<(n + 255) / 256, 256, 0, stream>>>(bufA, bufB, 32, 32, 128, n);
  }
  // dense: [32,32768] @ [32768,256], split-K=16 with atomic accumulation
  float* accb     = bufA;
  float* denseout = bufA + 16384;
  {
    int n = 32 * 256;
    snn_zero<<<(n + 255) / 256, 256, 0, stream>>>(accb, n);
  }
  snn_dense_splitk<<<dim3(32, 16), 32, 0, stream>>>(bufB, Wd1, accb,
                                                    32768, 256, 6000.0f);
  {
    int n = 32 * 256;
    snn_dense_finalize<<<(n + 255) / 256, 256, 0, stream>>>(accb, Dd1, denseout,
                                                            7500.0f, n);
  }
  // output layer -> d_out [32,10]
  snn_out_layer<<<1, 320, 0, stream>>>(denseout, Wout, Dout, (float*)d_out);
}